// Hbv4Dpl_40432822125189
// MI455X (gfx1250) — compile-verified
//
#include <hip/hip_runtime.h>
#include <stdint.h>

typedef uint32_t u32;
typedef __attribute__((ext_vector_type(4))) u32 v4u;
typedef __attribute__((ext_vector_type(8))) int  v8i;
typedef __attribute__((ext_vector_type(4))) int  v4i;

#define NSTEP 2000
#define NGRID 8192
#define TT    40               // timesteps per TDM tile
#define NT    (NSTEP / TT)     // 50 tiles
#define NBUF  3                // triple-buffered LDS pipeline
#define ROWF  (NGRID * 3)      // floats per full timestep row  = 24576
#define TROWF (32 * 3)         // floats per timestep per block = 96
#define TILEF (TT * TROWF)     // floats per LDS tile buffer    = 3840

// Issue one TDM 2D tile load: TT rows of TROWF dwords, row stride ROWF dwords.
// D# packing per CDNA5 ISA sec. 8.3/8.4 (group0 128b, group1 256b; groups 2/3 NULL).
__device__ __forceinline__ void tdm_load_tile(const float* gsrc, u32 lds_byte_off) {
  uint64_t ga = (uint64_t)(uintptr_t)gsrc;
  v4u g0;
  g0.x = 1u;                                                // count=1 (valid), no gather
  g0.y = lds_byte_off;                                      // lds_addr (bytes)
  g0.z = (u32)ga;                                           // global_addr[31:0]
  g0.w = ((u32)(ga >> 32) & 0x01FFFFFFu) | (2u << 30);      // global_addr[56:32] | type=2
  v8i g1;
  g1[0] = (int)(2u << 16);                                  // wg_mask=0, data_size=2 (4B)
  g1[1] = (int)((u32)TROWF << 16);                          // tensor_dim0[15:0]=96
  g1[2] = (int)((u32)NSTEP << 16);                          // dim0 hi=0 | tensor_dim1 lo=2000
  g1[3] = (int)((u32)TROWF << 16);                          // dim1 hi=0 | tile_dim0=96
  g1[4] = (int)TT;                                          // tile_dim1=40, tile_dim2=0
  g1[5] = (int)ROWF;                                        // tensor_dim0_stride lo = 24576
  g1[6] = 0;                                                // stride0 hi=0 | dim1_stride lo=0
  g1[7] = 0;
  v4i z4 = {0, 0, 0, 0};
  v8i z8 = {0, 0, 0, 0, 0, 0, 0, 0};
  __builtin_amdgcn_tensor_load_to_lds(g0, g1, z4, z4, z8, 0);
}

__global__ __launch_bounds__(32) void hbv_tdm_kernel(const float* __restrict__ x,
                                                     const float* __restrict__ p,
                                                     float* __restrict__ out) {
  __shared__ float lds[NBUF * TILEF];
  const int lane = threadIdx.x;                 // 0..31
  const int g    = blockIdx.x * 32 + lane;

  // ---- parameters: par = LO + p*(HI-LO) ----
  const float* pr = p + g * 14;
  float p0 = pr[0],  p1 = pr[1],  p2 = pr[2],  p3 = pr[3],  p4 = pr[4];
  float p5 = pr[5],  p6 = pr[6],  p7 = pr[7],  p8 = pr[8],  p9 = pr[9];
  float p10 = pr[10], p11 = pr[11], p12 = pr[12], p13 = pr[13];

  float beta  = 1.0f   + p0  * 5.0f;
  float fc    = 50.0f  + p1  * 950.0f;
  float k0    = 0.05f  + p2  * 0.85f;
  float k1    = 0.01f  + p3  * 0.49f;
  float k2    = 0.001f + p4  * 0.199f;
  float lp    = 0.2f   + p5  * 0.8f;
  float percp = p6 * 10.0f;
  float uzl   = p7 * 100.0f;
  float tt_   = -2.5f  + p8  * 5.0f;
  float cfmax = 0.5f   + p9  * 9.5f;
  float cfr   = p10 * 0.1f;
  float cwh   = p11 * 0.2f;
  float a_    = fmaxf(p12 * 2.9f, 0.0f) + 0.1f;
  float theta = fmaxf(p13 * 6.5f, 0.0f) + 0.5f;

  float inv_fc   = 1.0f / fc;
  float inv_lpfc = 1.0f / (lp * fc);
  float cfrcf    = cfr * cfmax;

  // ---- gamma-kernel conv weights; 1/(Gamma(a)*theta^a) cancels in normalization ----
  float w[15];
  {
    float am1 = a_ - 1.0f, invth = 1.0f / theta, s = 0.0f;
#pragma unroll
    for (int k = 0; k < 15; ++k) {
      float tk = (float)k + 0.5f;
      float u  = __expf(am1 * __logf(tk) - tk * invth);
      w[k] = u;
      s += u;
    }
    float invs = 1.0f / s;
#pragma unroll
    for (int k = 0; k < 15; ++k) w[k] *= invs;
  }

  // ---- state ----
  float snowpack = 0.001f, meltwater = 0.001f, sm = 0.001f, suz = 0.001f, slz = 0.001f;
  float h[14];
#pragma unroll
  for (int k = 0; k < 14; ++k) h[k] = 0.0f;

  const float* gx = x + (size_t)blockIdx.x * TROWF;     // this block's 96-dword column slab
  const u32 lds_base = (u32)(uintptr_t)(&lds[0]);

  // ---- prologue: prefetch tiles 0 and 1 (per-wave TDM, single-wave workgroup) ----
  tdm_load_tile(gx, lds_base);
  tdm_load_tile(gx + (size_t)TT * ROWF, lds_base + (u32)(TILEF * 4));

  for (int i = 0; i < NT; ++i) {
    // top-of-loop prefetch of tile i+2 into buf (i+2)%3, then wait for tile i.
    // TENSORcnt completes in-order, so wait<=2 guarantees the oldest (tile i) landed.
    if (i + 2 < NT) {
      // LDS reads of buf (i+2)%3 (done during iteration i-1) must retire first
      asm volatile("s_wait_dscnt 0" ::: "memory");
      tdm_load_tile(gx + (size_t)(i + 2) * TT * ROWF,
                    lds_base + (u32)(((i + 2) % NBUF) * TILEF * 4));
      __builtin_amdgcn_s_wait_tensorcnt(2);
    } else if (i + 1 < NT) {
      __builtin_amdgcn_s_wait_tensorcnt(1);
    } else {
      __builtin_amdgcn_s_wait_tensorcnt(0);
    }

    const float* xb = &lds[(i % NBUF) * TILEF + lane * 3];
    int tbase = i * TT;

#pragma unroll 5
    for (int st = 0; st < TT; ++st) {
      float precip = xb[0], potent = xb[1], tempre = xb[2];
      xb += TROWF;

      float dtt  = tempre - tt_;
      float rain = (dtt >= 0.0f) ? precip : 0.0f;
      float snow = precip - rain;
      snowpack += snow;
      float melt = fminf(fmaxf(cfmax * dtt, 0.0f), snowpack);
      meltwater += melt;
      snowpack  -= melt;
      float refr = fminf(fmaxf(-cfrcf * dtt, 0.0f), meltwater);
      snowpack  += refr;
      meltwater -= refr;
      float to_soil = fmaxf(meltwater - cwh * snowpack, 0.0f);
      meltwater -= to_soil;

      float ratio = sm * inv_fc;
      float sw    = fminf(__expf(beta * __logf(ratio)), 1.0f);  // >=0 always
      float rin   = rain + to_soil;
      float rech  = rin * sw;
      sm += rin - rech;
      float excess = fmaxf(sm - fc, 0.0f);
      sm -= excess;
      float ef = fminf(sm * inv_lpfc, 1.0f);
      float et = fminf(sm, potent * ef);
      sm = fmaxf(sm - et, 1e-5f);

      suz += rech + excess;
      float perc = fminf(suz, percp);
      suz -= perc;
      float q0 = k0 * fmaxf(suz - uzl, 0.0f);
      suz -= q0;
      float q1 = k1 * suz;
      suz -= q1;
      slz += perc;
      float q2 = k2 * slz;
      slz -= q2;
      float q = q0 + q1 + q2;

      // fused 15-tap causal conv (register ring)
      float o = w[0] * q;
#pragma unroll
      for (int k = 1; k < 15; ++k) o += w[k] * h[k - 1];
#pragma unroll
      for (int k = 13; k > 0; --k) h[k] = h[k - 1];
      h[0] = q;

      // output written once, never re-read: keep it out of L2 (NT store hint)
      __builtin_nontemporal_store(o, &out[(tbase + st) * NGRID + g]);
    }
  }
}

extern "C" void kernel_launch(void* const* d_in, const int* in_sizes, int n_in,
                              void* d_out, int out_size, void* d_ws, size_t ws_size,
                              hipStream_t stream) {
  const float* x = (const float*)d_in[0];        // (2000, 8192, 3) f32
  const float* p = (const float*)d_in[1];        // (8192, 14) f32
  float* out = (float*)d_out;                    // (2000, 8192, 1) f32
  hipLaunchKernelGGL(hbv_tdm_kernel, dim3(NGRID / 32), dim3(32), 0, stream, x, p, out);
}